// SGL_69234872811823
// MI455X (gfx1250) — compile-verified
//
#include <hip/hip_runtime.h>
#include <hip/hip_bf16.h>

#define N_NODES 50000
#define N_EDGES 800000
#define IN_DIM  64
#define HID     256
#define OUT_DIM 128

#define KC  64          // K-chunk staged in LDS
#define LDB (KC + 4)    // padded LDS row stride (floats) -> conflict-free ds_load_b64

typedef __attribute__((ext_vector_type(2))) float v2f;
typedef __attribute__((ext_vector_type(8))) float v8f;

// ---------------- degree / normalization ----------------

__global__ void SGL_zero_f32(float* __restrict__ p, int n) {
  int i = blockIdx.x * blockDim.x + threadIdx.x;
  if (i < n) p[i] = 0.0f;
}

__global__ void SGL_deg_kernel(const int* __restrict__ ei, float* __restrict__ deg) {
  int e = blockIdx.x * blockDim.x + threadIdx.x;
  if (e < N_EDGES) atomicAdd(&deg[ei[N_EDGES + e]], 1.0f);
}

__global__ void SGL_dis_kernel(float* __restrict__ deg) {
  int i = blockIdx.x * blockDim.x + threadIdx.x;
  if (i < N_NODES) {
    float d = deg[i];
    deg[i] = (d > 0.0f) ? rsqrtf(d) : 0.0f;
  }
}

// ---------------- fp32 WMMA GEMM, LDS-staged B panel ----------------
// C[m][col_off + n] = act( sum_k A[m][k] * B[k][n] + bias[n] )
// Block: 256 threads = 8 waves. Block tile = 128 rows x 64 cols.
// Wave w owns m-tile (mb*128 + w*16); computes 16x64 via 4 accumulators.
// B panel (KC x 64) staged in LDS transposed: Bs[n][k], row stride LDB=KC+4
//   -> lane l15 reads Bs[(j*16+l15)*LDB + k + 2*half .. +1] as ds_load_b64,
//      banks 4*l15 + {0..3}: all 64 banks covered, conflict-free.
// WMMA f32 16x16x4 per-lane layout:
//   A: lane L -> M=L&15, VGPR{0,1} = K = k + 2*(L>>4) + {0,1}
//   B: lane L -> N=L&15, VGPR{0,1} = K = k + 2*(L>>4) + {0,1}
//   D: VGPR v -> M = v + 8*(L>>4), N = L&15
__global__ void __launch_bounds__(256)
SGL_gemm_wmma_f32(const float* __restrict__ A, const float* __restrict__ B,
                  const float* __restrict__ bias, float* __restrict__ C,
                  int M, int K, int Nn, int ldc, int col_off, int relu_flag) {
  __shared__ float Bs[64 * LDB];

  const int tid  = threadIdx.x;
  const int lane = tid & 31;
  const int w    = tid >> 5;               // wave 0..7
  const int nblocks = Nn >> 6;             // # of 64-col blocks
  const int nb = blockIdx.x % nblocks;
  const int mb = blockIdx.x / nblocks;
  const int half = lane >> 4;              // 0 = lanes 0..15, 1 = lanes 16..31
  const int l15  = lane & 15;

  const int mtile   = mb * 128 + w * 16;   // this wave's m-tile base
  const int colBase = nb * 64;

  int rowA = mtile + l15;
  if (rowA > M - 1) rowA = M - 1;          // tail waves: clamp reads, guard stores later
  const float* __restrict__ Arow = A + (size_t)rowA * K + 2 * half;

  // staging role: thread t loads column (t&63) of the 64-wide panel
  const int nstage = tid & 63;
  const int kstage = tid >> 6;             // 0..3

  v8f acc0 = {}, acc1 = {}, acc2 = {}, acc3 = {};

  for (int kc = 0; kc < K; kc += KC) {
    // cooperative stage: B[kc+k][colBase+n] -> Bs[n*LDB + k] (coalesced global b32)
#pragma unroll
    for (int i = 0; i < KC / 4; ++i) {
      int k = kstage + 4 * i;
      Bs[nstage * LDB + k] = B[(size_t)(kc + k) * Nn + colBase + nstage];
    }
    __syncthreads();

    const float* __restrict__ Ap = Arow + kc;
    if (kc + KC < K) __builtin_prefetch(Ap + KC, 0, 0);   // global_prefetch_b8

#pragma unroll 4
    for (int k = 0; k < KC; k += 4) {
      v2f a = *(const v2f*)(Ap + k);                      // global_load_b64
      const int kb = k + 2 * half;
      v2f b0 = *(const v2f*)&Bs[(0 * 16 + l15) * LDB + kb];   // ds_load_b64 x4
      v2f b1 = *(const v2f*)&Bs[(1 * 16 + l15) * LDB + kb];
      v2f b2 = *(const v2f*)&Bs[(2 * 16 + l15) * LDB + kb];
      v2f b3 = *(const v2f*)&Bs[(3 * 16 + l15) * LDB + kb];
      acc0 = __builtin_amdgcn_wmma_f32_16x16x4_f32(false, a, false, b0, (short)0, acc0, false, false);
      acc1 = __builtin_amdgcn_wmma_f32_16x16x4_f32(false, a, false, b1, (short)0, acc1, false, false);
      acc2 = __builtin_amdgcn_wmma_f32_16x16x4_f32(false, a, false, b2, (short)0, acc2, false, false);
      acc3 = __builtin_amdgcn_wmma_f32_16x16x4_f32(false, a, false, b3, (short)0, acc3, false, false);
    }
    __syncthreads();   // protect LDS before next chunk overwrites
  }

  if (mtile >= M) return;   // after final barrier: tail waves exit (whole-tile granularity)

  v8f accs[4] = {acc0, acc1, acc2, acc3};
#pragma unroll
  for (int j = 0; j < 4; ++j) {
    const int col = colBase + j * 16 + l15;
    const float bval = bias ? bias[col] : 0.0f;
#pragma unroll
    for (int v = 0; v < 8; ++v) {
      int m = mtile + v + 8 * half;
      float val = accs[j][v] + bval;
      if (relu_flag) val = fmaxf(val, 0.0f);
      C[(size_t)m * ldc + col_off + col] = val;
    }
  }
}

// ---------------- aggregation (scatter-add) ----------------

// x3[col][256 + c] += norm(e) * T1[row][c]   (one block per edge, thread = channel)
__global__ void SGL_scatter_hid(const int* __restrict__ ei, const float* __restrict__ dis,
                                const float* __restrict__ T1, float* __restrict__ x3) {
  int e = blockIdx.x;
  int c = threadIdx.x;                       // 0..255
  int row = ei[e];
  int col = ei[N_EDGES + e];
  float norm = dis[row] * dis[col];
  float v = T1[(size_t)row * HID + c] * norm;
  atomicAdd(&x3[(size_t)col * (2 * HID) + HID + c], v);
}

// agg2[col][c] += norm(e) * T2[row][c]   (one block per edge, thread = channel)
__global__ void SGL_scatter_out(const int* __restrict__ ei, const float* __restrict__ dis,
                                const float* __restrict__ T2, float* __restrict__ agg2) {
  int e = blockIdx.x;
  int c = threadIdx.x;                       // 0..127
  int row = ei[e];
  int col = ei[N_EDGES + e];
  float norm = dis[row] * dis[col];
  float v = T2[(size_t)row * OUT_DIM + c] * norm;
  atomicAdd(&agg2[(size_t)col * OUT_DIM + c], v);
}

__global__ void SGL_init_agg1(const float* __restrict__ b, float* __restrict__ x3) {
  int idx = blockIdx.x * blockDim.x + threadIdx.x;    // over N*HID
  if (idx < N_NODES * HID) {
    int n = idx >> 8;
    int c = idx & 255;
    x3[(size_t)n * (2 * HID) + HID + c] = b[c];
  }
}

__global__ void SGL_relu_agg1(float* __restrict__ x3) {
  int idx = blockIdx.x * blockDim.x + threadIdx.x;    // over N*HID
  if (idx < N_NODES * HID) {
    int n = idx >> 8;
    int c = idx & 255;
    size_t o = (size_t)n * (2 * HID) + HID + c;
    x3[o] = fmaxf(x3[o], 0.0f);
  }
}

__global__ void SGL_init_agg2(const float* __restrict__ b, float* __restrict__ agg2) {
  int idx = blockIdx.x * blockDim.x + threadIdx.x;    // over N*OUT_DIM
  if (idx < N_NODES * OUT_DIM) {
    int c = idx & (OUT_DIM - 1);
    agg2[idx] = b[c];
  }
}

// x7 = x5 + relu(agg2)  (conv2 bias already folded into agg2 init)
__global__ void SGL_combine(const float* __restrict__ X5, const float* __restrict__ agg2,
                            float* __restrict__ X7) {
  int i = blockIdx.x * blockDim.x + threadIdx.x;
  if (i < N_NODES * OUT_DIM) X7[i] = X5[i] + fmaxf(agg2[i], 0.0f);
}

// ---------------- final head: two 128-dots per node (wave per node) ----------------

__global__ void SGL_node_head(const float* __restrict__ X7, const float* __restrict__ w_fc2,
                              const float* __restrict__ b_fc2, const float* __restrict__ w_conv3,
                              const float* __restrict__ b_conv3, float* __restrict__ out,
                              float* __restrict__ u) {
  int wave = blockIdx.x * (blockDim.x >> 5) + (threadIdx.x >> 5);
  int lane = threadIdx.x & 31;
  if (wave >= N_NODES) return;
  const float* xr = X7 + (size_t)wave * OUT_DIM;
  float t = 0.0f, s = 0.0f;
#pragma unroll
  for (int c = lane; c < OUT_DIM; c += 32) {
    float xv = xr[c];
    t += xv * w_fc2[c];
    s += xv * w_conv3[c];
  }
#pragma unroll
  for (int off = 16; off > 0; off >>= 1) {
    t += __shfl_down(t, off, 32);
    s += __shfl_down(s, off, 32);
  }
  if (lane == 0) {
    out[wave] = t + b_fc2[0] + b_conv3[0];  // fc2 bias + conv3 bias pre-added
    u[wave]   = s;                          // per-node conv3 message value
  }
}

__global__ void SGL_scatter_final(const int* __restrict__ ei, const float* __restrict__ dis,
                                  const float* __restrict__ u, float* __restrict__ out) {
  int e = blockIdx.x * blockDim.x + threadIdx.x;
  if (e < N_EDGES) {
    int row = ei[e];
    int col = ei[N_EDGES + e];
    float norm = dis[row] * dis[col];
    atomicAdd(&out[col], norm * u[row]);
  }
}

// ---------------- launch ----------------

extern "C" void kernel_launch(void* const* d_in, const int* in_sizes, int n_in,
                              void* d_out, int out_size, void* d_ws, size_t ws_size,
                              hipStream_t stream) {
  const float* x       = (const float*)d_in[0];
  const int*   ei      = (const int*)  d_in[1];   // [2*E] flat: row then col
  const float* w_fc    = (const float*)d_in[2];
  const float* b_fc    = (const float*)d_in[3];
  const float* w_conv1 = (const float*)d_in[4];
  const float* b_conv1 = (const float*)d_in[5];
  const float* w_fc1   = (const float*)d_in[6];
  const float* b_fc1   = (const float*)d_in[7];
  const float* w_conv2 = (const float*)d_in[8];
  const float* b_conv2 = (const float*)d_in[9];
  const float* w_fc2   = (const float*)d_in[10];
  const float* b_fc2   = (const float*)d_in[11];
  const float* w_conv3 = (const float*)d_in[12];
  const float* b_conv3 = (const float*)d_in[13];
  float* out = (float*)d_out;
  float* ws  = (float*)d_ws;

  // workspace layout (floats); regions reused once their producer/consumers retire
  float* dis  = ws;                               // 50 048
  float* x3   = dis + 50048;                      // 50 000 * 512
  float* T1   = x3 + (size_t)N_NODES * 2 * HID;   // 50 000 * 256 (xw1; later X5 + agg2)
  float* X5   = T1;                               // 50 000 * 128
  float* agg2 = T1 + (size_t)N_NODES * OUT_DIM;   // 50 000 * 128
  float* T2   = T1 + (size_t)N_NODES * HID;       // 50 000 * 128 (xw2; later X7)
  float* X7   = T2;                               // 50 000 * 128
  float* u    = T2 + (size_t)N_NODES * OUT_DIM;   // 50 000

  const int waves_per_block = 256 / 32;
  const int mblocks = (N_NODES + 127) / 128;      // 391 (last block: 5 live m-tiles)

  // 1) symmetric normalization factors
  SGL_zero_f32<<<(N_NODES + 255) / 256, 256, 0, stream>>>(dis, N_NODES);
  SGL_deg_kernel<<<(N_EDGES + 255) / 256, 256, 0, stream>>>(ei, dis);
  SGL_dis_kernel<<<(N_NODES + 255) / 256, 256, 0, stream>>>(dis);

  // 2) x1 = relu(x @ w_fc + b_fc) -> x3[:, 0:256]
  SGL_gemm_wmma_f32<<<mblocks * (HID / 64), 256, 0, stream>>>(
      x, w_fc, b_fc, x3, N_NODES, IN_DIM, HID, 2 * HID, 0, 1);

  // 3) T1 = x @ w_conv1 (transform before propagate)
  SGL_gemm_wmma_f32<<<mblocks * (HID / 64), 256, 0, stream>>>(
      x, w_conv1, nullptr, T1, N_NODES, IN_DIM, HID, HID, 0, 0);

  // 4) x3[:, 256:512] = relu(seg_sum(norm * T1[row]) + b_conv1)
  SGL_init_agg1<<<(N_NODES * HID + 255) / 256, 256, 0, stream>>>(b_conv1, x3);
  SGL_scatter_hid<<<N_EDGES, HID, 0, stream>>>(ei, dis, T1, x3);
  SGL_relu_agg1<<<(N_NODES * HID + 255) / 256, 256, 0, stream>>>(x3);

  // 5) X5 = relu(x3 @ w_fc1 + b_fc1)
  SGL_gemm_wmma_f32<<<mblocks * (OUT_DIM / 64), 256, 0, stream>>>(
      x3, w_fc1, b_fc1, X5, N_NODES, 2 * HID, OUT_DIM, OUT_DIM, 0, 1);

  // 6) T2 = x3 @ w_conv2
  SGL_gemm_wmma_f32<<<mblocks * (OUT_DIM / 64), 256, 0, stream>>>(
      x3, w_conv2, nullptr, T2, N_NODES, 2 * HID, OUT_DIM, OUT_DIM, 0, 0);

  // 7) agg2 = seg_sum(norm * T2[row]) + b_conv2 ; X7 = X5 + relu(agg2)
  SGL_init_agg2<<<(N_NODES * OUT_DIM + 255) / 256, 256, 0, stream>>>(b_conv2, agg2);
  SGL_scatter_out<<<N_EDGES, OUT_DIM, 0, stream>>>(ei, dis, T2, agg2);
  SGL_combine<<<(N_NODES * OUT_DIM + 255) / 256, 256, 0, stream>>>(X5, agg2, X7);

  // 8) head: out = x7@w_fc2 + b_fc2 + b_conv3 + seg_sum(norm * (x7@w_conv3)[row])
  SGL_node_head<<<(N_NODES + waves_per_block - 1) / waves_per_block, 256, 0, stream>>>(
      X7, w_fc2, b_fc2, w_conv3, b_conv3, out, u);
  SGL_scatter_final<<<(N_EDGES + 255) / 256, 256, 0, stream>>>(ei, dis, u, out);
}